// SoftSplat_85667417686079
// MI455X (gfx1250) — compile-verified
//
#include <hip/hip_runtime.h>
#include <cstdint>

#if defined(__has_builtin)
#  if __has_builtin(__builtin_amdgcn_global_load_async_to_lds_b32)
#    define USE_ASYNC_LDS 1
#  endif
#  if __has_builtin(__builtin_amdgcn_s_wait_asynccnt)
#    define HAVE_WAIT_ASYNC 1
#  endif
#endif

#define BATCH 4
#define H1 256
#define W1 448
#define NPIX1 (H1 * W1)
#define H2 128
#define W2 224
#define NPIX2 (H2 * W2)
#define H3 64
#define W3 112
#define NPIX3 (H3 * W3)
#define TPB 256

// ---- workspace layout (float offsets) ----
#define WS_MET 0                                   // [2][B][NPIX1] metrics m12,m21
#define WS_DEN1 (2 * BATCH * NPIX1)                // [2][B][NPIX1]
#define WS_DEN2 (WS_DEN1 + 2 * BATCH * NPIX1)      // [2][B][NPIX2]
#define WS_DEN3 (WS_DEN2 + 2 * BATCH * NPIX2)      // [2][B][NPIX3]

typedef __attribute__((address_space(1))) int as1_int;
typedef __attribute__((address_space(3))) int as3_int;

__device__ __forceinline__ void atomAddF(float* p, float v) {
  // no-return f32 atomic -> global_atomic_add_f32 (STOREcnt, fire-and-forget)
  __hip_atomic_fetch_add(p, v, __ATOMIC_RELAXED, __HIP_MEMORY_SCOPE_AGENT);
}

__device__ __forceinline__ void async_cp_b32(const void* gsrc, void* ldst) {
#ifdef USE_ASYNC_LDS
  __builtin_amdgcn_global_load_async_to_lds_b32(
      (as1_int*)(uintptr_t)gsrc,
      (as3_int*)(uint32_t)(uintptr_t)ldst,
      0, 0);
#else
  *(float*)ldst = *(const float*)gsrc;
#endif
}

__device__ __forceinline__ void wait_async_zero() {
#ifdef USE_ASYNC_LDS
#  ifdef HAVE_WAIT_ASYNC
  __builtin_amdgcn_s_wait_asynccnt(0);
#  else
  asm volatile("s_wait_asynccnt 0" ::: "memory");
#  endif
#endif
}

// ---------------------------------------------------------------------------
// Kernel 1: occlusion metric  m = mean_c |imgA - backwarp(imgB, flow)|
// grid: (NPIX1/TPB, BATCH, 2)   dir 0: m12, dir 1: m21
// imgA tile is staged to LDS with the gfx1250 async DMA path while the
// data-dependent imgB gather is issued -> copy/compute overlap.
// ---------------------------------------------------------------------------
__global__ void metric_kernel(const float* __restrict__ img1,
                              const float* __restrict__ img2,
                              const float* __restrict__ f12,
                              const float* __restrict__ f21,
                              float* __restrict__ met) {
  __shared__ float sA[3 * TPB];
  const int tid = threadIdx.x;
  const int p = blockIdx.x * TPB + tid;
  const int b = blockIdx.y;
  const int dir = blockIdx.z;

  const float* imgA = dir ? img2 : img1;
  const float* imgB = dir ? img1 : img2;
  const float* flow = dir ? f21 : f12;

  // stage imgA (3 channels, this pixel tile) into LDS asynchronously
  const float* ga = imgA + (size_t)b * 3 * NPIX1 + p;
  async_cp_b32(ga, &sA[tid]);
  async_cp_b32(ga + NPIX1, &sA[TPB + tid]);
  async_cp_b32(ga + 2 * NPIX1, &sA[2 * TPB + tid]);

  const int x = p % W1;
  const int y = p / W1;
  const float f0 = flow[((size_t)b * 2 + 0) * NPIX1 + p];
  const float f1 = flow[((size_t)b * 2 + 1) * NPIX1 + p];

  // grid_sample coords (align_corners=True, base grid at -1+1/W .. 1-1/W)
  const float gx = (-1.0f + (2.0f * x + 1.0f) / (float)W1) + f0 * (2.0f / (W1 - 1.0f));
  const float gy = (-1.0f + (2.0f * y + 1.0f) / (float)H1) + f1 * (2.0f / (H1 - 1.0f));
  const float px = (gx + 1.0f) * 0.5f * (W1 - 1.0f);
  const float py = (gy + 1.0f) * 0.5f * (H1 - 1.0f);

  const int x0 = (int)floorf(px);
  const int y0 = (int)floorf(py);
  const float ax = px - (float)x0;
  const float ay = py - (float)y0;

  float acc0 = 0.f, acc1 = 0.f, acc2 = 0.f;
  const float* ib = imgB + (size_t)b * 3 * NPIX1;
#pragma unroll
  for (int dy = 0; dy < 2; ++dy) {
#pragma unroll
    for (int dx = 0; dx < 2; ++dx) {
      const int xi = x0 + dx;
      const int yi = y0 + dy;
      const float w = (dx ? ax : 1.0f - ax) * (dy ? ay : 1.0f - ay);
      if (xi >= 0 && xi < W1 && yi >= 0 && yi < H1) {
        const int idx = yi * W1 + xi;
        acc0 += ib[idx] * w;
        acc1 += ib[NPIX1 + idx] * w;
        acc2 += ib[2 * NPIX1 + idx] * w;
      }
    }
  }

  wait_async_zero();
  __syncthreads();

  const float m = (fabsf(sA[tid] - acc0) + fabsf(sA[TPB + tid] - acc1) +
                   fabsf(sA[2 * TPB + tid] - acc2)) * (1.0f / 3.0f);
  met[((size_t)dir * BATCH + b) * NPIX1 + p] = m;
}

// ---------------------------------------------------------------------------
// Kernel 2: softmax-splat scatter-accumulate for one pyramid level.
// Pyramid flow/metric resize (bilinear, align_corners) is folded in on the
// fly by linearity:  fl[l] = resize(flow) * fscale,  mt[l] = resize(m).
// grid: (NP/TPB, BATCH, 2)
// ---------------------------------------------------------------------------
__global__ void splat_kernel(const float* __restrict__ flowA,
                             const float* __restrict__ flowB,
                             const float* __restrict__ met,
                             const float* __restrict__ featA,
                             const float* __restrict__ featB,
                             const float* __restrict__ alpha,
                             float* __restrict__ num,   // level base in d_out
                             float* __restrict__ den,   // [2][B][NP]
                             int C, int Ctot, int c0A, int c0B,
                             int Hl, int Wl, float fscale, int accDen) {
  const int tid = threadIdx.x;
  const int p = blockIdx.x * TPB + tid;
  const int b = blockIdx.y;
  const int dir = blockIdx.z;
  const int NP = Hl * Wl;
  const int x = p % Wl;
  const int y = p / Wl;

  const float* flow = dir ? flowB : flowA;
  const float* feat = dir ? featB : featA;
  const int c0 = dir ? c0B : c0A;

  // bilinear sample of full-res flow & metric at this level's coordinates
  const float ry = (float)y * ((H1 - 1.0f) / (float)(Hl - 1));
  const float rx = (float)x * ((W1 - 1.0f) / (float)(Wl - 1));
  int ry0 = (int)floorf(ry);
  int rx0 = (int)floorf(rx);
  const float wy = ry - (float)ry0;
  const float wx = rx - (float)rx0;
  const int ry1 = min(ry0 + 1, H1 - 1);
  const int rx1 = min(rx0 + 1, W1 - 1);
  const float w00 = (1.f - wy) * (1.f - wx), w01 = (1.f - wy) * wx;
  const float w10 = wy * (1.f - wx), w11 = wy * wx;
  const int i00 = ry0 * W1 + rx0, i01 = ry0 * W1 + rx1;
  const int i10 = ry1 * W1 + rx0, i11 = ry1 * W1 + rx1;

  const float* fb = flow + (size_t)b * 2 * NPIX1;
  const float flx = (fb[i00] * w00 + fb[i01] * w01 + fb[i10] * w10 + fb[i11] * w11) * fscale;
  const float* fby = fb + NPIX1;
  const float fly = (fby[i00] * w00 + fby[i01] * w01 + fby[i10] * w10 + fby[i11] * w11) * fscale;
  const float* mp = met + ((size_t)dir * BATCH + b) * NPIX1;
  const float m = mp[i00] * w00 + mp[i01] * w01 + mp[i10] * w10 + mp[i11] * w11;

  const float wgt = __expf(alpha[0] * m);

  // splat target taps
  const float gx = (float)x + flx;
  const float gy = (float)y + fly;
  const int tx0 = (int)floorf(gx);
  const int ty0 = (int)floorf(gy);
  const float axx = gx - (float)tx0;
  const float ayy = gy - (float)ty0;

  int ti[4];
  float tw[4];
  bool tv[4];
#pragma unroll
  for (int k = 0; k < 4; ++k) {
    const int dx = k & 1, dy = k >> 1;
    const int xi = tx0 + dx, yi = ty0 + dy;
    tv[k] = (xi >= 0) && (xi < Wl) && (yi >= 0) && (yi < Hl);
    ti[k] = tv[k] ? (yi * Wl + xi) : 0;
    tw[k] = (dx ? axx : 1.0f - axx) * (dy ? ayy : 1.0f - ayy);
  }

  if (accDen) {
    float* dp = den + ((size_t)dir * BATCH + b) * NP;
#pragma unroll
    for (int k = 0; k < 4; ++k)
      if (tv[k]) atomAddF(dp + ti[k], wgt * tw[k]);
  }

  float* nb = num + ((size_t)b * Ctot + c0) * NP;
  const float* fp = feat + (size_t)b * C * NP + p;
  for (int c = 0; c < C; ++c) {
    if (c + 4 < C) __builtin_prefetch(fp + (size_t)(c + 4) * NP, 0, 0);
    const float v = fp[(size_t)c * NP] * wgt;
    float* o = nb + (size_t)c * NP;
#pragma unroll
    for (int k = 0; k < 4; ++k)
      if (tv[k]) atomAddF(o + ti[k], v * tw[k]);
  }
}

// ---------------------------------------------------------------------------
// Kernel 3: normalize  out[c] = num[c] / (den + 1e-7), fully 128-bit
// vectorized (b128 loads/stores); den reciprocals held in registers across
// the whole channel loop (reuse factor = Ctot).
// grid: (NP/4/TPB, BATCH); NP divisible by 1024 at every level.
// ---------------------------------------------------------------------------
__global__ void norm_kernel(float* __restrict__ out, const float* __restrict__ den,
                            int Ctot, int Chalf, int NP) {
  const int q = blockIdx.x * TPB + threadIdx.x;  // float4 index within plane
  const int b = blockIdx.y;
  const int NQ = NP >> 2;

  const float4 d0 = ((const float4*)(den + ((size_t)0 * BATCH + b) * NP))[q];
  const float4 d1 = ((const float4*)(den + ((size_t)1 * BATCH + b) * NP))[q];
  const float4 r0 = make_float4(1.0f / (d0.x + 1e-7f), 1.0f / (d0.y + 1e-7f),
                                1.0f / (d0.z + 1e-7f), 1.0f / (d0.w + 1e-7f));
  const float4 r1 = make_float4(1.0f / (d1.x + 1e-7f), 1.0f / (d1.y + 1e-7f),
                                1.0f / (d1.z + 1e-7f), 1.0f / (d1.w + 1e-7f));

  float4* ob = (float4*)(out + (size_t)b * Ctot * NP) + q;
  for (int c = 0; c < Ctot; ++c) {
    const float4 r = (c < Chalf) ? r0 : r1;
    float4 v = ob[(size_t)c * NQ];
    v.x *= r.x; v.y *= r.y; v.z *= r.z; v.w *= r.w;
    ob[(size_t)c * NQ] = v;
  }
}

extern "C" void kernel_launch(void* const* d_in, const int* in_sizes, int n_in,
                              void* d_out, int out_size, void* d_ws, size_t ws_size,
                              hipStream_t stream) {
  (void)in_sizes; (void)n_in; (void)ws_size;
  const float* img1 = (const float*)d_in[0];
  const float* img2 = (const float*)d_in[1];
  const float* f1_1 = (const float*)d_in[2];
  const float* f1_2 = (const float*)d_in[3];
  const float* f1_3 = (const float*)d_in[4];
  const float* f2_1 = (const float*)d_in[5];
  const float* f2_2 = (const float*)d_in[6];
  const float* f2_3 = (const float*)d_in[7];
  const float* fl12 = (const float*)d_in[8];
  const float* fl21 = (const float*)d_in[9];
  const float* alpha = (const float*)d_in[10];

  float* out = (float*)d_out;
  float* ws = (float*)d_ws;
  float* met = ws + WS_MET;
  float* den1 = ws + WS_DEN1;
  float* den2 = ws + WS_DEN2;
  float* den3 = ws + WS_DEN3;

  const size_t O2 = (size_t)BATCH * 70 * NPIX1;
  const size_t O3 = O2 + (size_t)BATCH * 128 * NPIX2;

  // zero accumulators (graph-capturable)
  (void)hipMemsetAsync(d_out, 0, (size_t)out_size * sizeof(float), stream);
  (void)hipMemsetAsync(den1, 0,
                       (size_t)(2 * BATCH) * (NPIX1 + NPIX2 + NPIX3) * sizeof(float), stream);

  const dim3 blk(TPB);

  metric_kernel<<<dim3(NPIX1 / TPB, BATCH, 2), blk, 0, stream>>>(img1, img2, fl12, fl21, met);

  // Level 1: images (3ch, accumulates shared denominator) + features (32ch)
  splat_kernel<<<dim3(NPIX1 / TPB, BATCH, 2), blk, 0, stream>>>(
      fl12, fl21, met, img1, img2, alpha, out, den1, 3, 70, 0, 35, H1, W1, 0.5f, 1);
  splat_kernel<<<dim3(NPIX1 / TPB, BATCH, 2), blk, 0, stream>>>(
      fl12, fl21, met, f1_1, f2_1, alpha, out, den1, 32, 70, 3, 38, H1, W1, 0.5f, 0);
  // Level 2
  splat_kernel<<<dim3(NPIX2 / TPB, BATCH, 2), blk, 0, stream>>>(
      fl12, fl21, met, f1_2, f2_2, alpha, out + O2, den2, 64, 128, 0, 64, H2, W2, 0.25f, 1);
  // Level 3
  splat_kernel<<<dim3(NPIX3 / TPB, BATCH, 2), blk, 0, stream>>>(
      fl12, fl21, met, f1_3, f2_3, alpha, out + O3, den3, 96, 192, 0, 96, H3, W3, 0.125f, 1);

  norm_kernel<<<dim3(NPIX1 / 4 / TPB, BATCH), blk, 0, stream>>>(out, den1, 70, 35, NPIX1);
  norm_kernel<<<dim3(NPIX2 / 4 / TPB, BATCH), blk, 0, stream>>>(out + O2, den2, 128, 64, NPIX2);
  norm_kernel<<<dim3(NPIX3 / 4 / TPB, BATCH), blk, 0, stream>>>(out + O3, den3, 192, 96, NPIX3);
}